// NeuralNetwork_51479478010284
// MI455X (gfx1250) — compile-verified
//
#include <hip/hip_runtime.h>
#include <math.h>

typedef float v2f __attribute__((ext_vector_type(2)));
typedef float v8f __attribute__((ext_vector_type(8)));

#define BATCH   16384
#define NTOT    1024
#define NIN     128
#define NOUT    64
#define HID_END 960   // NTOT - NOUT
#define REFRAC  0.9f

// ---------------------------------------------------------------------------
// Pack kernel: Wt[n][k] = W[k][960+n] * (0.9 for hidden rows, 1.0 otherwise)
// 64 x 1024 fp32 = 256 KB, L2-resident thereafter.
// ---------------------------------------------------------------------------
__global__ void pack_weights_kernel(const float* __restrict__ W,
                                    float* __restrict__ Wt) {
  int idx = blockIdx.x * blockDim.x + threadIdx.x;   // 0 .. 65535
  int n = idx >> 10;                                 // 0 .. 63
  int k = idx & (NTOT - 1);                          // 0 .. 1023
  float scale = (k >= NIN && k < HID_END) ? REFRAC : 1.0f;
  Wt[idx] = W[(size_t)k * NTOT + HID_END + n] * scale;
}

// ---------------------------------------------------------------------------
// D = A*B + C with fp32 WMMA 16x16x4
// ---------------------------------------------------------------------------
static __device__ __forceinline__ v8f wmma_k4(v2f a, v2f b, v8f c) {
  return __builtin_amdgcn_wmma_f32_16x16x4_f32(
      /*neg_a=*/false, a, /*neg_b=*/false, b,
      /*c_mod=*/(short)0, c, /*reuse_a=*/false, /*reuse_b=*/false);
}

static __device__ __forceinline__ float activate(float x, int id) {
  if (id == 0) return x > 0.0f ? x : 0.0f;         // relu
  if (id == 1) return tanhf(x);                    // tanh
  if (id == 2) return 1.0f / (1.0f + expf(-x));    // sigmoid
  return x;                                        // identity
}

// ---------------------------------------------------------------------------
// Fused GEMM + bias + heterogeneous activation.
// One wave -> 16 rows x all 64 output cols (4 WMMA N-tiles).
// Block = 256 threads = 8 waves = 128 rows. Grid = 16384/128 = 128 blocks.
// ---------------------------------------------------------------------------
__global__ void __launch_bounds__(256)
fused_out_kernel(const float* __restrict__ prev,   // [B, 1024]
                 const float* __restrict__ inp,    // [B, 128]
                 const float* __restrict__ Wt,     // [64, 1024] packed
                 const float* __restrict__ bias,   // [1024]
                 const int*   __restrict__ acts,   // [1024]
                 float* __restrict__ out)          // [B, 64]
{
  const int lane = threadIdx.x & 31;
  const int wave = threadIdx.x >> 5;
  const int ln15 = lane & 15;          // M (A) / N (B,C,D) within tile
  const int hi   = lane >> 4;          // lane half
  const int kb   = hi * 2;             // per-lane K sub-offset in a K=4 chunk
  const int m0   = (blockIdx.x * 8 + wave) * 16;

  // Per-lane B pointers for the 4 N-tiles (Wt row-major, stride 1024)
  const float* bp0 = Wt + (size_t)(ln15 +  0) * NTOT + kb;
  const float* bp1 = Wt + (size_t)(ln15 + 16) * NTOT + kb;
  const float* bp2 = Wt + (size_t)(ln15 + 32) * NTOT + kb;
  const float* bp3 = Wt + (size_t)(ln15 + 48) * NTOT + kb;

  v8f acc0 = {0.f,0.f,0.f,0.f,0.f,0.f,0.f,0.f};
  v8f acc1 = acc0, acc2 = acc0, acc3 = acc0;

  // K in [0, 128): A comes from input_values (row stride 128)
  {
    const float* ap = inp + (size_t)(m0 + ln15) * NIN + kb;
    #pragma unroll 8
    for (int k = 0; k < NIN; k += 4) {
      v2f a = *(const v2f*)(ap + k);
      acc0 = wmma_k4(a, *(const v2f*)(bp0 + k), acc0);
      acc1 = wmma_k4(a, *(const v2f*)(bp1 + k), acc1);
      acc2 = wmma_k4(a, *(const v2f*)(bp2 + k), acc2);
      acc3 = wmma_k4(a, *(const v2f*)(bp3 + k), acc3);
    }
  }
  // K in [128, 1024): A comes from prev_states (refractory folded into Wt)
  {
    const float* ap = prev + (size_t)(m0 + ln15) * NTOT + kb;
    #pragma unroll 8
    for (int k = NIN; k < NTOT; k += 4) {
      v2f a = *(const v2f*)(ap + k);
      acc0 = wmma_k4(a, *(const v2f*)(bp0 + k), acc0);
      acc1 = wmma_k4(a, *(const v2f*)(bp1 + k), acc1);
      acc2 = wmma_k4(a, *(const v2f*)(bp2 + k), acc2);
      acc3 = wmma_k4(a, *(const v2f*)(bp3 + k), acc3);
    }
  }

  // Epilogue: C/D layout -> lane holds column (16t + ln15), rows m0+8*hi+v.
  const int mtop = m0 + hi * 8;
  {
    const int n = 0 * 16 + ln15;
    const float bt = bias[HID_END + n];
    const int   id = acts[HID_END + n];
    #pragma unroll
    for (int v = 0; v < 8; ++v)
      out[(size_t)(mtop + v) * NOUT + n] = activate(acc0[v] + bt, id);
  }
  {
    const int n = 1 * 16 + ln15;
    const float bt = bias[HID_END + n];
    const int   id = acts[HID_END + n];
    #pragma unroll
    for (int v = 0; v < 8; ++v)
      out[(size_t)(mtop + v) * NOUT + n] = activate(acc1[v] + bt, id);
  }
  {
    const int n = 2 * 16 + ln15;
    const float bt = bias[HID_END + n];
    const int   id = acts[HID_END + n];
    #pragma unroll
    for (int v = 0; v < 8; ++v)
      out[(size_t)(mtop + v) * NOUT + n] = activate(acc2[v] + bt, id);
  }
  {
    const int n = 3 * 16 + ln15;
    const float bt = bias[HID_END + n];
    const int   id = acts[HID_END + n];
    #pragma unroll
    for (int v = 0; v < 8; ++v)
      out[(size_t)(mtop + v) * NOUT + n] = activate(acc3[v] + bt, id);
  }
}

// ---------------------------------------------------------------------------
extern "C" void kernel_launch(void* const* d_in, const int* in_sizes, int n_in,
                              void* d_out, int out_size, void* d_ws, size_t ws_size,
                              hipStream_t stream) {
  const float* prev = (const float*)d_in[0];   // [16384,1024]
  const float* inp  = (const float*)d_in[1];   // [16384,128]
  const float* W    = (const float*)d_in[2];   // [1024,1024]
  const float* bias = (const float*)d_in[3];   // [1024]
  const int*   acts = (const int*)  d_in[4];   // [1024]
  float* out = (float*)d_out;                  // [16384,64]
  float* Wt  = (float*)d_ws;                   // 64*1024 floats = 256 KB scratch

  (void)in_sizes; (void)n_in; (void)out_size; (void)ws_size;

  // 1) Pack/transpose/scale the live 1024x64 weight slice.
  pack_weights_kernel<<<(NOUT * NTOT) / 256, 256, 0, stream>>>(W, Wt);

  // 2) Fused GEMM + bias + activation over the output slice.
  fused_out_kernel<<<BATCH / 128, 256, 0, stream>>>(prev, inp, Wt, bias, acts, out);
}